// MobileNet_90452011254295
// MI455X (gfx1250) — compile-verified
//
#include <hip/hip_runtime.h>
#include <hip/hip_bf16.h>

// ---------------------------------------------------------------------------
// MobileNetV1 (batch 1, fp32) for gfx1250 / MI455X.
//
// - stem conv3x3 s2 + BN + ReLU         : direct fused kernel
// - 13x [ depthwise3x3 + BN + ReLU      : direct fused kernel
//         pointwise1x1 (stride) +BN+ReLU: WMMA F32 16x16x4 GEMM,
//             B panel (cin x 16 activations) staged ONCE per N-tile into LDS
//             via gfx1250 async-to-LDS loads (ASYNCcnt) when available,
//             shared by 8 waves each computing a different 16-row M-tile ]
// - global avgpool(7x7) + FC(1024->100) : single fused kernel
//
// WMMA f32 16x16x4 lane mapping (CDNA5 ISA 7.12.2, 32-bit operands):
//   A (16x4, MxK): lane<16 -> {K=k+0,k+1}, lane>=16 -> {K=k+2,k+3}, M=lane&15
//   B (4x16, KxN): vgpr r = row K=r (lanes 0-15) / K=r+2 (lanes 16-31), N=lane&15
//   C (16x16):     vgpr r -> M = r + 8*(lane>=16), N = lane&15
// ---------------------------------------------------------------------------

typedef float v2f __attribute__((ext_vector_type(2)));
typedef float v8f __attribute__((ext_vector_type(8)));

typedef __attribute__((address_space(1))) int as1_int;   // global ("__device__")
typedef __attribute__((address_space(3))) int as3_int;   // LDS ("__shared__")

#define EPSBN 1e-5f

// ---- gfx1250 async global->LDS path (guarded so the build never breaks) ----
#if defined(__has_builtin)
#if __has_builtin(__builtin_amdgcn_global_load_async_to_lds_b32)
#define HAVE_ASYNC_TO_LDS 1
#endif
#if __has_builtin(__builtin_amdgcn_s_wait_asynccnt)
#define HAVE_WAIT_ASYNC 1
#endif
#endif

__device__ __forceinline__ void async_fill_b32(const float* gp, float* lp) {
#ifdef HAVE_ASYNC_TO_LDS
    // signature (probe-confirmed by clang diagnostic):
    //   (AS1 int* src, AS3 int* dst, imm int offset, imm int cpol)
    __builtin_amdgcn_global_load_async_to_lds_b32(
        (as1_int*)(float*)gp, (as3_int*)lp, 0, 0);
#else
    *lp = *gp;                       // fallback: global load + ds_store
#endif
}

__device__ __forceinline__ void async_fill_wait() {
#ifdef HAVE_ASYNC_TO_LDS
#ifdef HAVE_WAIT_ASYNC
    __builtin_amdgcn_s_wait_asynccnt(0);
#else
    asm volatile("s_wait_asynccnt 0x0" ::: "memory");
#endif
#endif
}

// ---------------- stem: conv3x3 s2 p1 (3->32) + BN + ReLU -------------------
__global__ __launch_bounds__(256) void stem_conv_kernel(
    const float* __restrict__ in,    // [3,224,224]
    const float* __restrict__ w,     // [32,3,3,3]
    const float* __restrict__ g, const float* __restrict__ bta,
    const float* __restrict__ mu, const float* __restrict__ var,
    float* __restrict__ out)         // [32,112,112]
{
    const int HW = 112 * 112;
    int idx = blockIdx.x * blockDim.x + threadIdx.x;
    if (idx >= 32 * HW) return;
    int co = idx / HW;
    int p  = idx - co * HW;
    int y  = p / 112;
    int x  = p - y * 112;

    float s = 0.f;
    for (int ci = 0; ci < 3; ++ci) {
        const float* ip = in + ci * 224 * 224;
        const float* wp = w + (co * 3 + ci) * 9;
        #pragma unroll
        for (int ky = 0; ky < 3; ++ky) {
            int iy = 2 * y + ky - 1;
            if (iy < 0 || iy >= 224) continue;
            #pragma unroll
            for (int kx = 0; kx < 3; ++kx) {
                int ix = 2 * x + kx - 1;
                if (ix < 0 || ix >= 224) continue;
                s += ip[iy * 224 + ix] * wp[ky * 3 + kx];
            }
        }
    }
    float iv = g[co] * rsqrtf(var[co] + EPSBN);
    float sh = bta[co] - mu[co] * iv;
    out[idx] = fmaxf(s * iv + sh, 0.f);
}

// ---------------- depthwise 3x3 s1 p1 + BN + ReLU ---------------------------
__global__ __launch_bounds__(256) void dw_conv_kernel(
    const float* __restrict__ in,    // [C,H,W]
    const float* __restrict__ w,     // [C,1,3,3]
    const float* __restrict__ g, const float* __restrict__ bta,
    const float* __restrict__ mu, const float* __restrict__ var,
    float* __restrict__ out,         // [C,H,W]
    int C, int H, int W)
{
    int idx = blockIdx.x * blockDim.x + threadIdx.x;
    int HW = H * W;
    if (idx >= C * HW) return;
    int c = idx / HW;
    int p = idx - c * HW;
    int y = p / W;
    int x = p - y * W;

    const float* ip = in + (size_t)c * HW;
    const float* wp = w + c * 9;
    float s = 0.f;
    #pragma unroll
    for (int ky = 0; ky < 3; ++ky) {
        int iy = y + ky - 1;
        if (iy < 0 || iy >= H) continue;
        #pragma unroll
        for (int kx = 0; kx < 3; ++kx) {
            int ix = x + kx - 1;
            if (ix < 0 || ix >= W) continue;
            s += ip[iy * W + ix] * wp[ky * 3 + kx];
        }
    }
    float iv = g[c] * rsqrtf(var[c] + EPSBN);
    float sh = bta[c] - mu[c] * iv;
    out[idx] = fmaxf(s * iv + sh, 0.f);
}

// ---------------- pointwise 1x1 (stride s) as WMMA f32 GEMM + BN + ReLU -----
// Grid: (tilesN, ceil(tilesM/8)); block = 256 threads = 8 waves.
// Workgroup stages B panel bpanel[cin][16] into LDS once (async-to-LDS when
// available), then each wave computes one 16x16 C tile for its M-tile.
// No divergence around WMMA (EXEC all ones); edge columns use clamped load
// addresses and predicated stores.
__global__ __launch_bounds__(256) void pw_wmma_kernel(
    const float* __restrict__ act,   // [cin,H,W]  (full-res dw output)
    const float* __restrict__ w,     // [cout,cin]
    const float* __restrict__ g, const float* __restrict__ bta,
    const float* __restrict__ mu, const float* __restrict__ var,
    float* __restrict__ out,         // [cout,Ho,Wo]
    int cin, int cout, int H, int W, int stride)
{
    extern __shared__ float bpanel[];          // [cin][16], <= 64 KB

    const int Wo = W / stride;
    const int Ho = H / stride;
    const int P  = Ho * Wo;
    const int HW = H * W;

    const int tid  = threadIdx.x;
    const int wave = tid >> 5;
    const int lane = tid & 31;
    const int half = lane >> 4;     // 0: lanes 0-15, 1: lanes 16-31
    const int lidx = lane & 15;
    const int tn   = blockIdx.x;    // one N-tile per workgroup

    // ---- cooperative B-panel fill: bpanel[k][n] = act[k, col(n)] ----
    {
        int nn   = tid & 15;                       // invariant across e-steps
        int col  = tn * 16 + nn;
        int colc = (col < P) ? col : (P - 1);      // clamp ragged edge
        int oy = colc / Wo;
        int ox = colc - oy * Wo;
        size_t coff = (size_t)(oy * stride) * W + (size_t)(ox * stride);
        int total = cin * 16;                      // always a multiple of 256
        for (int e = tid; e < total; e += 256) {
            int kk = e >> 4;
            async_fill_b32(act + (size_t)kk * HW + coff, &bpanel[e]);
        }
        async_fill_wait();
        __syncthreads();
    }

    // ---- per-wave 16x16 tile: M-tile = blockIdx.y*8 + wave ----
    const int tilesM = cout >> 4;
    int mt = blockIdx.y * 8 + wave;
    const bool mValid = (mt < tilesM);
    if (!mValid) mt = tilesM - 1;                  // clamp: compute, no store

    const float* wp = w + (size_t)(mt * 16 + lidx) * cin + (half << 1);
    __builtin_prefetch(wp, 0, 0);                  // global_prefetch_b8

    v8f acc = {};
    for (int k = 0; k < cin; k += 16) {
        // 4 A fragments (b64 each) -- streaming weights
        v2f a0 = *(const v2f*)(wp + k);
        v2f a1 = *(const v2f*)(wp + k + 4);
        v2f a2 = *(const v2f*)(wp + k + 8);
        v2f a3 = *(const v2f*)(wp + k + 12);
        // 8 B scalars from LDS (bank-conflict-free: lanes 0-15 and 16-31
        // land on disjoint bank groups)
        const float* bp = &bpanel[(k + (half << 1)) * 16 + lidx];
        v2f b0, b1, b2, b3;
        b0.x = bp[0];   b0.y = bp[16];
        b1.x = bp[64];  b1.y = bp[80];
        b2.x = bp[128]; b2.y = bp[144];
        b3.x = bp[192]; b3.y = bp[208];
        acc = __builtin_amdgcn_wmma_f32_16x16x4_f32(false, a0, false, b0, (short)0, acc, false, false);
        acc = __builtin_amdgcn_wmma_f32_16x16x4_f32(false, a1, false, b1, (short)0, acc, false, false);
        acc = __builtin_amdgcn_wmma_f32_16x16x4_f32(false, a2, false, b2, (short)0, acc, false, false);
        acc = __builtin_amdgcn_wmma_f32_16x16x4_f32(false, a3, false, b3, (short)0, acc, false, false);
    }

    // ---- fused BN + ReLU store ----
    int col = tn * 16 + lidx;
    if (mValid && col < P) {
        #pragma unroll
        for (int r = 0; r < 8; ++r) {
            int mrow = mt * 16 + r + (half << 3);
            float iv = g[mrow] * rsqrtf(var[mrow] + EPSBN);
            float sh = bta[mrow] - mu[mrow] * iv;
            out[(size_t)mrow * P + col] = fmaxf(acc[r] * iv + sh, 0.f);
        }
    }
}

// ---------------- global avgpool 7x7 + FC 1024->100 -------------------------
__global__ __launch_bounds__(256) void head_kernel(
    const float* __restrict__ feat,  // [1024,7,7]
    const float* __restrict__ fw,    // [100,1024]
    const float* __restrict__ fb,    // [100]
    float* __restrict__ out)         // [100]
{
    __shared__ float gvec[1024];
    for (int c = threadIdx.x; c < 1024; c += 256) {
        float s = 0.f;
        #pragma unroll
        for (int i = 0; i < 49; ++i) s += feat[c * 49 + i];
        gvec[c] = s * (1.0f / 49.0f);
    }
    __syncthreads();
    for (int j = threadIdx.x; j < 100; j += 256) {
        float s = fb[j];
        for (int c = 0; c < 1024; ++c) s += gvec[c] * fw[j * 1024 + c];
        out[j] = s;
    }
}

// ---------------------------------------------------------------------------
// Input layout (setup_inputs() insertion order, leaves flattened):
//   0: x  1: stem_w  2..5: stem_bn(gamma,beta,mean,var)
//   per block b (b=0..12), base = 6 + 10*b:
//     +0 dw_w  +1..+4 dw_bn(g,b,m,v)  +5 pw_w  +6..+9 pw_bn(g,b,m,v)
//   136: fc_w  137: fc_b
// ---------------------------------------------------------------------------
extern "C" void kernel_launch(void* const* d_in, const int* in_sizes, int n_in,
                              void* d_out, int out_size, void* d_ws, size_t ws_size,
                              hipStream_t stream)
{
    static const int CFGC[13][3] = {
        {32, 64, 1},  {64, 128, 2},  {128, 128, 1}, {128, 256, 2}, {256, 256, 1},
        {256, 512, 2},{512, 512, 1}, {512, 512, 1}, {512, 512, 1}, {512, 512, 1},
        {512, 512, 1},{512, 1024, 2},{1024, 1024, 1}};

    const float* x      = (const float*)d_in[0];
    const float* stem_w = (const float*)d_in[1];
    const float* stem_g = (const float*)d_in[2];
    const float* stem_b = (const float*)d_in[3];
    const float* stem_m = (const float*)d_in[4];
    const float* stem_v = (const float*)d_in[5];
    const float* fc_w   = (const float*)d_in[6 + 13 * 10];
    const float* fc_b   = (const float*)d_in[6 + 13 * 10 + 1];
    float* out = (float*)d_out;

    // Ping-pong activation buffers in workspace (max tensor: 64*112*112 floats)
    const size_t BUF = (size_t)64 * 112 * 112;   // 802816 floats = 3.2 MB
    float* bufA = (float*)d_ws;
    float* bufB = bufA + BUF;

    // stem -> bufA [32,112,112]
    {
        int n = 32 * 112 * 112;
        stem_conv_kernel<<<(n + 255) / 256, 256, 0, stream>>>(
            x, stem_w, stem_g, stem_b, stem_m, stem_v, bufA);
    }

    int H = 112, W = 112;
    for (int bi = 0; bi < 13; ++bi) {
        const int cin = CFGC[bi][0], cout = CFGC[bi][1], s = CFGC[bi][2];
        const float* const* B = (const float* const*)(d_in + 6 + bi * 10);

        // depthwise: bufA -> bufB  [cin,H,W]
        {
            int n = cin * H * W;
            dw_conv_kernel<<<(n + 255) / 256, 256, 0, stream>>>(
                bufA, B[0], B[1], B[2], B[3], B[4], bufB, cin, H, W);
        }

        // pointwise (strided) WMMA GEMM with LDS-staged B: bufB -> bufA
        {
            int Ho = H / s, Wo = W / s, P = Ho * Wo;
            int tilesN = (P + 15) >> 4;
            int tilesM = cout >> 4;
            dim3 grid(tilesN, (tilesM + 7) >> 3);
            size_t shmem = (size_t)cin * 16 * sizeof(float);
            pw_wmma_kernel<<<grid, 256, shmem, stream>>>(
                bufB, B[5], B[6], B[7], B[8], B[9], bufA, cin, cout, H, W, s);
        }
        H /= s; W /= s;
    }

    // avgpool + FC: bufA [1024,7,7] -> out [100]
    head_kernel<<<1, 256, 0, stream>>>(bufA, fc_w, fc_b, out);
}